// DistanceTransform_71983651881343
// MI455X (gfx1250) — compile-verified
//
#include <hip/hip_runtime.h>

// ---- CDNA5 vector types ----
typedef float        v8f  __attribute__((ext_vector_type(8)));
typedef float        v2f  __attribute__((ext_vector_type(2)));
typedef unsigned int v4u  __attribute__((ext_vector_type(4)));
typedef int          v4i  __attribute__((ext_vector_type(4)));
typedef int          v8i  __attribute__((ext_vector_type(8)));

#define HW     160
#define XS     164              // LDS row stride in floats (coprime-ish with 64 banks)
#define NPX    (HW * HW)        // 25600
#define TPB    1024             // 32 wave32 waves on one WGP
#define PXT    (NPX / TPB)      // 25 pixels per thread (exact)
#define NTILES 100              // 10x10 grid of 16x16 WMMA tiles
#define NWAVES (TPB / 32)

__launch_bounds__(TPB, 1)
__global__ void dt_kernel(const float* __restrict__ img,
                          float* __restrict__ out)
{
    __shared__ float Xs[HW * XS];    // boundary plane (evolving)
    __shared__ float T2s[HW * XS];   // per-iteration horizontal pass H2(X)
    __shared__ int   sflag;

    const int tid = threadIdx.x;
    const float* src = img + (size_t)blockIdx.x * NPX;
    float*       dst = out + (size_t)blockIdx.x * NPX;

    // ---------------- TDM bulk load: plane -> LDS (stride XS) ----------------
    // Iterate mode: 160 tiles of one 160-float row each; lds_addr_increment=XS
    // lays rows out at stride 164 for bank-conflict-free access.
    if (tid < 32) {
        unsigned int       ldsa = (unsigned int)(unsigned long long)(&Xs[0]);
        unsigned long long ga   = (unsigned long long)src;
        v4u g0 = { 1u,                                  // count=1 (valid D#)
                   ldsa,                                // lds_addr (bytes)
                   (unsigned int)(ga & 0xffffffffu),    // global_addr[31:0]
                   (unsigned int)((ga >> 32) & 0x1ffffffu) | (2u << 30) }; // type=2
        v8i g1 = { (int)((2u << 16) | (1u << 19)),      // data_size=4B, iterate_enable
                   (int)(160u << 16),                   // tensor_dim0 = 160
                   (int)(160u << 16),                   // tensor_dim1 = 160
                   (int)(160u << 16),                   // tile_dim0   = 160
                   1,                                   // tile_dim1   = 1
                   160,                                 // tensor_dim0_stride
                   (int)(160u << 16),                   // tensor_dim1_stride
                   0 };
        v4i g2 = { 1,                                   // tensor_dim2
                   XS,                                  // lds_addr_increment
                   160,                                 // global_addr_increment
                   (int)(159u << 16) };                 // iterate_count: 160 rows
        v4i g3 = { 0, 0, 0, 0 };
        v8i gx = { 0, 0, 0, 0, 0, 0, 0, 0 };            // surplus arg in clang-23 builtin
        __builtin_amdgcn_tensor_load_to_lds(g0, g1, g2, g3, gx, 0);
        __builtin_amdgcn_s_wait_tensorcnt(0);
        if (tid == 0) sflag = -1;
    }
    __syncthreads();

    // ---------------- constants / lane decomposition ----------------
    const float h  = 0.35f;
    const float wa = __expf(-1.0f / h);            // edge weight  e^{-1/h}
    const float wb = __expf(-1.41421356237f / h);  // corner       e^{-sqrt2/h}

    const int lane = tid & 31;
    const int lr   = lane & 15;   // matrix row (A) / column (B,C,D)
    const int lhi  = lane >> 4;   // half-wave select
    const int wid  = tid >> 5;

    // B operand for T2 = X * M2, M2 = tridiag(wb, wa, wb), 16x16, split into
    // four 4x16 K-chunks. Layout mirrors the A K-pattern: vgpr0 holds K-rows
    // {4kc+2*lhi}, vgpr1 holds {4kc+2*lhi+1}.
    v2f bch[4];
#pragma unroll
    for (int kc = 0; kc < 4; ++kc) {
        const int k0 = kc * 4 + 2 * lhi;
        const int k1 = k0 + 1;
        bch[kc].x = (k0 == lr) ? wa : ((k0 == lr + 1 || k0 + 1 == lr) ? wb : 0.0f);
        bch[kc].y = (k1 == lr) ? wa : ((k1 == lr + 1 || k1 + 1 == lr) ? wb : 0.0f);
    }

    float acc_out[PXT];
#pragma unroll
    for (int j = 0; j < PXT; ++j) acc_out[j] = 0.0f;

    // ---------------- 160 wavefront iterations, fully LDS-resident ----------------
    for (int it = 0; it < HW; ++it) {
        // ---- pass A: T2 = H2(X) per 16x16 tile via chained WMMA ----
        for (int t = wid; t < NTILES; t += NWAVES) {
            const int r0 = (t / 10) * 16, c0 = (t % 10) * 16;
            v8f acc = { 0.0f, 0.0f, 0.0f, 0.0f, 0.0f, 0.0f, 0.0f, 0.0f };
#pragma unroll
            for (int kc = 0; kc < 4; ++kc) {
                const int col = c0 + kc * 4 + 2 * lhi;
                const float* xr = &Xs[(r0 + lr) * XS + col];
                v2f a; a.x = xr[0]; a.y = xr[1];          // ds_load_b64, 64 banks hit
                acc = __builtin_amdgcn_wmma_f32_16x16x4_f32(
                          false, a, false, bch[kc], (short)0, acc, false, false);
            }
            // replicate-edge single-column fixups (outside the 16-wide band)
            const int colL = (c0 == 0) ? 0 : (c0 - 1);
            const int colR = (c0 + 16 > HW - 1) ? (HW - 1) : (c0 + 16);
            union { v8f v; float f[8]; } U; U.v = acc;
#pragma unroll
            for (int v = 0; v < 8; ++v) {
                const int m = r0 + v + 8 * lhi;           // D-layout row
                float t2 = U.f[v];
                if (lr == 0)  t2 += wb * Xs[m * XS + colL];
                if (lr == 15) t2 += wb * Xs[m * XS + colR];
                T2s[m * XS + c0 + lr] = t2;
            }
        }
        __syncthreads();

        // ---- pass B: vertical combine + H1 row conv, cdt/mask/accumulate ----
        const float offs = (float)((3 * it) >> 1);
        unsigned int mmask = 0u;
#pragma unroll
        for (int j = 0; j < PXT; ++j) {
            const int p  = tid + (j << 10);
            const int y  = p / HW;
            const int x  = p - y * HW;
            const int xm = (x == 0) ? 0 : x - 1;
            const int xp = (x == HW - 1) ? x : x + 1;
            const int ym = (y == 0) ? 0 : y - 1;
            const int yp = (y == HW - 1) ? y : y + 1;
            const float t1 = wa * (Xs[y * XS + xm] + Xs[y * XS + xp]) + Xs[y * XS + x];
            const float u  = t1 + T2s[ym * XS + x] + T2s[yp * XS + x];
            const float cdt = (u > 0.0f) ? (-h * __logf(u)) : 0.0f;
            if (cdt > 0.0f) { acc_out[j] += offs + cdt; mmask |= (1u << j); }
        }
        __syncthreads();

        // ---- write phase: advance the front; iteration-stamped change flag ----
        if (mmask) {
#pragma unroll
            for (int j = 0; j < PXT; ++j) {
                if (mmask & (1u << j)) {
                    const int p = tid + (j << 10);
                    const int y = p / HW;
                    Xs[y * XS + (p - y * HW)] = 1.0f;
                }
            }
            sflag = it;   // benign race: all writers store the same value
        }
        __syncthreads();
        if (sflag != it) break;   // exact: remaining iterations are no-ops
    }

    // ---------------- write result (coalesced) ----------------
#pragma unroll
    for (int j = 0; j < PXT; ++j)
        dst[tid + (j << 10)] = acc_out[j];
}

extern "C" void kernel_launch(void* const* d_in, const int* in_sizes, int n_in,
                              void* d_out, int out_size, void* d_ws, size_t ws_size,
                              hipStream_t stream)
{
    (void)n_in; (void)d_ws; (void)ws_size; (void)out_size;
    const float* img = (const float*)d_in[0];
    float* out = (float*)d_out;
    const int planes = in_sizes[0] / NPX;   // B*C = 24 -> one WGP-resident block each
    dt_kernel<<<planes, TPB, 0, stream>>>(img, out);
}